// HyperNetwork_9990093931021
// MI455X (gfx1250) — compile-verified
//
#include <hip/hip_runtime.h>

// HyperNetwork: per-channel GEMM (256x256x64, f32) via V_WMMA_F32_16X16X4_F32,
// then tiny shared 64->9 projection. Memory-bound (~172MB @ 23.3TB/s), so we
// keep full fp32 precision on the matrix pipe.

typedef __attribute__((ext_vector_type(2))) float v2f;
typedef __attribute__((ext_vector_type(4))) float v4f;
typedef __attribute__((ext_vector_type(8))) float v8f;

#define B_DIM 256
#define C_DIM 512
#define NZ    256
#define ZD    64
#define KOUT  9

#define LDS_STRIDE 260   // 256 + 4 floats pad: keeps 16B alignment, kills bank conflicts
#define A_TILE_STRIDE 68 // 64 + 4 pad for the activation tile

__global__ __launch_bounds__(128)
void hypernet_wmma_f32_kernel(const float* __restrict__ z,
                              const float* __restrict__ W_in,
                              const float* __restrict__ b_in,
                              const float* __restrict__ W_out,
                              const float* __restrict__ b_out,
                              float* __restrict__ out)
{
    __shared__ float lds[16 * LDS_STRIDE]; // 16.6 KB; reused for the 16x64 a-tile later

    const int c      = blockIdx.y;        // channel 0..511
    const int m_base = blockIdx.x * 16;   // batch-row tile base
    const int tid    = threadIdx.x;

    // ---- Stage A tile: x[m_base..m_base+15, c, 0..255] into LDS (float4 loads) ----
    {
        #pragma unroll
        for (int i = 0; i < 8; ++i) {
            int id   = tid + 128 * i;     // 1024 float4 slots: 16 rows x 64 quads
            int row  = id >> 6;
            int col4 = id & 63;
            const float* src =
                z + (((size_t)(m_base + row) * C_DIM + c) * NZ) + (size_t)col4 * 4;
            v4f v = *(const v4f*)src;
            *(v4f*)&lds[row * LDS_STRIDE + col4 * 4] = v;
        }
    }
    __syncthreads();

    const int wave   = tid >> 5;          // 0..3 : each wave owns 16 z_dim columns
    const int lane   = tid & 31;
    const int half   = lane >> 4;         // lane half selects K pairs (0,1) vs (2,3)
    const int l16    = lane & 15;
    const int n_col  = wave * 16 + l16;   // z_dim column 0..63

    const float* Wc = W_in + (size_t)c * NZ * ZD;

    // ---- 64 chained V_WMMA_F32_16X16X4_F32 over K=256 ----
    v8f acc = {};
    #pragma unroll 8
    for (int k = 0; k < NZ; k += 4) {
        const int K0 = k + 2 * half;
        // A fragment (16x4): lane holds row l16, K0/K0+1 -> one ds_load_b64
        v2f a = *(const v2f*)&lds[l16 * LDS_STRIDE + K0];
        // B fragment (4x16): lane holds column n_col, rows K0/K0+1
        v2f b;
        b.x = Wc[(size_t)K0 * ZD + n_col];
        b.y = Wc[(size_t)(K0 + 1) * ZD + n_col];
        // 8 args: (neg_a, A, neg_b, B, c_mod, C, reuse_a, reuse_b)
        acc = __builtin_amdgcn_wmma_f32_16x16x4_f32(
                  false, a, false, b, (short)0, acc, false, false);
    }

    // ---- add b_in[c, n_col] to every row of this column ----
    const float bias = b_in[(size_t)c * ZD + n_col];
    #pragma unroll
    for (int i = 0; i < 8; ++i) acc[i] += bias;

    __syncthreads(); // all waves done reading staged A; reuse LDS for a-tile

    // C/D layout: VGPR i, lanes 0-15 -> M=i, lanes 16-31 -> M=i+8; N = n_col
    #pragma unroll
    for (int i = 0; i < 8; ++i) {
        const int m_local = half ? (i + 8) : i;
        lds[m_local * A_TILE_STRIDE + n_col] = acc[i];
    }
    __syncthreads();

    // ---- second (tiny) GEMM: [16 x 64] @ [64 x 9] + b_out -> out[B, C, 9] ----
    for (int idx = tid; idx < 16 * KOUT; idx += 128) {
        const int r = idx / KOUT;
        const int j = idx - r * KOUT;
        float s = b_out[j];
        #pragma unroll
        for (int zz = 0; zz < ZD; ++zz)
            s += lds[r * A_TILE_STRIDE + zz] * W_out[zz * KOUT + j];
        out[(((size_t)(m_base + r)) * C_DIM + c) * KOUT + j] = s;
    }
}

extern "C" void kernel_launch(void* const* d_in, const int* in_sizes, int n_in,
                              void* d_out, int out_size, void* d_ws, size_t ws_size,
                              hipStream_t stream) {
    const float* z     = (const float*)d_in[0];
    const float* W_in  = (const float*)d_in[1];
    const float* b_in  = (const float*)d_in[2];
    const float* W_out = (const float*)d_in[3];
    const float* b_out = (const float*)d_in[4];
    float* out = (float*)d_out;

    dim3 grid(B_DIM / 16, C_DIM); // 16 row-tiles x 512 channels = 8192 WGs
    hypernet_wmma_f32_kernel<<<grid, 128, 0, stream>>>(z, W_in, b_in, W_out, b_out, out);
}